// OhemCELoss_34402688041014
// MI455X (gfx1250) — compile-verified
//
#include <hip/hip_runtime.h>
#include <hip/hip_bf16.h>

// ---------------------------------------------------------------------------
// OHEM cross-entropy loss for logits[16,19,512,512] fp32, labels[16,512,512].
// Stage 1: per-pixel log-softmax NLL, 4 pixels/thread with float4 (b128)
//          loads -- one pass over 318MB of logits, pure HBM-bound (~13.7us
//          at 23.3 TB/s). No index divisions (2D grid: blockIdx.y = batch).
// Stage 2: exact top-k selection via 4-pass radix-select on float bit
//          patterns over the 16MB loss array (L2-resident: 192MB L2).
// Stage 3: sum of losses above cutoff (+ exact tie correction) / k.
// Wave32 reduction of the final sum goes through V_WMMA_F32_16X16X4_F32.
// ---------------------------------------------------------------------------

#define NCLASS 19
#define HW_CONST (512 * 512)
#define NBATCH 16
#define THRESH_NEGLOG 0.35667494393873245f /* -log(0.7) */
#define IGNORE_LB 255

typedef __attribute__((ext_vector_type(2))) float v2f;
typedef __attribute__((ext_vector_type(8))) float v8f;

// meta layout (32-bit words), placed in d_ws right after the loss array:
//  [0] validCount  [1] hardCount  [2] k  [3] prefix (cutoff bits)
//  [4] remaining   [5] countAbove [6] sumAbove (float bits) [7] pad
//  [8..263] hist[256]

__global__ void ohem_init(unsigned* __restrict__ meta) {
  unsigned t = threadIdx.x;
  if (t < 8) meta[t] = 0u;
  meta[8 + t] = 0u; // blockDim.x == 256 -> zero hist
}

// grid: (HW/(256*4), NBATCH), block: 256. Each thread: 4 consecutive pixels.
__global__ void ohem_loss(const float* __restrict__ logits,
                          const int* __restrict__ labels,
                          float* __restrict__ loss,
                          unsigned* __restrict__ meta) {
  const int n = blockIdx.y;
  const int hw = (blockIdx.x * blockDim.x + threadIdx.x) * 4;
  const float* base = logits + (size_t)n * NCLASS * HW_CONST + hw;
  const int* labp = labels + (size_t)n * HW_CONST + hw;
  // hint the prefetcher at the label stream ahead of us (global_prefetch_b8)
  __builtin_prefetch(labp + 16384, 0, 0);

  float4 v[NCLASS];
#pragma unroll
  for (int c = 0; c < NCLASS; ++c)
    v[c] = *(const float4*)(base + (size_t)c * HW_CONST);

  float4 m = v[0];
#pragma unroll
  for (int c = 1; c < NCLASS; ++c) {
    m.x = fmaxf(m.x, v[c].x);
    m.y = fmaxf(m.y, v[c].y);
    m.z = fmaxf(m.z, v[c].z);
    m.w = fmaxf(m.w, v[c].w);
  }
  float4 s = make_float4(0.f, 0.f, 0.f, 0.f);
#pragma unroll
  for (int c = 0; c < NCLASS; ++c) {
    s.x += __expf(v[c].x - m.x);
    s.y += __expf(v[c].y - m.y);
    s.z += __expf(v[c].z - m.z);
    s.w += __expf(v[c].w - m.w);
  }
  float4 lse = make_float4(m.x + __logf(s.x), m.y + __logf(s.y),
                           m.z + __logf(s.z), m.w + __logf(s.w));

  int4 lab = *(const int4*)labp;
  float4 l = make_float4(0.f, 0.f, 0.f, 0.f);
  int isValid = 0, isHard = 0;
  if (lab.x != IGNORE_LB) {
    isValid++;
    l.x = lse.x - base[(size_t)lab.x * HW_CONST + 0]; // cache-hot reload
    if (l.x > THRESH_NEGLOG) isHard++;
  }
  if (lab.y != IGNORE_LB) {
    isValid++;
    l.y = lse.y - base[(size_t)lab.y * HW_CONST + 1];
    if (l.y > THRESH_NEGLOG) isHard++;
  }
  if (lab.z != IGNORE_LB) {
    isValid++;
    l.z = lse.z - base[(size_t)lab.z * HW_CONST + 2];
    if (l.z > THRESH_NEGLOG) isHard++;
  }
  if (lab.w != IGNORE_LB) {
    isValid++;
    l.w = lse.w - base[(size_t)lab.w * HW_CONST + 3];
    if (l.w > THRESH_NEGLOG) isHard++;
  }
  *(float4*)(loss + (size_t)n * HW_CONST + hw) = l;

  // wave32 reduction of the two counters
  for (int o = 16; o > 0; o >>= 1) {
    isValid += __shfl_xor(isValid, o, 32);
    isHard += __shfl_xor(isHard, o, 32);
  }
  if ((threadIdx.x & 31) == 0) {
    if (isValid) atomicAdd(&meta[0], (unsigned)isValid);
    if (isHard) atomicAdd(&meta[1], (unsigned)isHard);
  }
}

__global__ void ohem_compute_k(unsigned* __restrict__ meta) {
  if (threadIdx.x == 0 && blockIdx.x == 0) {
    unsigned n_min = meta[0] / 16u;
    unsigned n_hard = meta[1];
    unsigned k = n_hard > n_min ? n_hard : n_min;
    meta[2] = k;              // k
    meta[3] = 0u;             // prefix
    meta[4] = k;              // remaining
    meta[5] = 0u;             // countAbove
    ((float*)meta)[6] = 0.0f; // sumAbove
  }
}

__global__ void ohem_hist(const float* __restrict__ loss, int P,
                          const unsigned* __restrict__ meta,
                          unsigned* __restrict__ hist, int bytePos) {
  __shared__ unsigned sh[256];
  sh[threadIdx.x] = 0u;
  __syncthreads();
  const unsigned prefix = meta[3];
  const unsigned maskHi =
      (bytePos == 3) ? 0u : (0xFFFFFFFFu << ((bytePos + 1) * 8));
  const unsigned pfx = prefix & maskHi;
  const int P4 = P >> 2;
  const int stride = gridDim.x * blockDim.x;
  const float4* loss4 = (const float4*)loss;
  for (int p = blockIdx.x * blockDim.x + threadIdx.x; p < P4; p += stride) {
    float4 v = loss4[p];
    unsigned k0 = __float_as_uint(v.x), k1 = __float_as_uint(v.y);
    unsigned k2 = __float_as_uint(v.z), k3 = __float_as_uint(v.w);
    if ((k0 & maskHi) == pfx) atomicAdd(&sh[(k0 >> (bytePos * 8)) & 0xFFu], 1u);
    if ((k1 & maskHi) == pfx) atomicAdd(&sh[(k1 >> (bytePos * 8)) & 0xFFu], 1u);
    if ((k2 & maskHi) == pfx) atomicAdd(&sh[(k2 >> (bytePos * 8)) & 0xFFu], 1u);
    if ((k3 & maskHi) == pfx) atomicAdd(&sh[(k3 >> (bytePos * 8)) & 0xFFu], 1u);
  }
  __syncthreads();
  unsigned c = sh[threadIdx.x];
  if (c) atomicAdd(&hist[threadIdx.x], c);
}

__global__ void ohem_select(unsigned* __restrict__ meta,
                            unsigned* __restrict__ hist, int bytePos) {
  __shared__ unsigned sh[256];
  sh[threadIdx.x] = hist[threadIdx.x];
  __syncthreads();
  if (threadIdx.x == 0) {
    unsigned remaining = meta[4];
    unsigned cum = 0;
    int sel = 0;
    for (int b = 255; b >= 0; --b) {
      unsigned nc = cum + sh[b];
      if (nc >= remaining) {
        sel = b;
        meta[4] = remaining - cum;
        break;
      }
      cum = nc;
    }
    meta[3] |= ((unsigned)sel) << (bytePos * 8);
  }
  __syncthreads();
  hist[threadIdx.x] = 0u; // ready for next pass / next graph replay
}

__global__ void ohem_sum(const float* __restrict__ loss, int P,
                         unsigned* __restrict__ meta) {
  const unsigned cutoff = meta[3];
  float part = 0.0f;
  int cnt = 0;
  const int P4 = P >> 2;
  const int stride = gridDim.x * blockDim.x;
  const float4* loss4 = (const float4*)loss;
  for (int p = blockIdx.x * blockDim.x + threadIdx.x; p < P4; p += stride) {
    float4 v = loss4[p];
    if (__float_as_uint(v.x) > cutoff) { part += v.x; cnt++; }
    if (__float_as_uint(v.y) > cutoff) { part += v.y; cnt++; }
    if (__float_as_uint(v.z) > cutoff) { part += v.z; cnt++; }
    if (__float_as_uint(v.w) > cutoff) { part += v.w; cnt++; }
  }
  // ---- wave32 sum via V_WMMA_F32_16X16X4_F32 ----
  // A (16x4 f32, 2 VGPRs): lane L -> A[L%16][K], K in {0,2}/{1,3}.
  // partial in a.x, 0 in a.y; B = ones(4x16) -> D[m][n] = rowsum(m).
  // Executed unconditionally by all 256 threads: EXEC all 1s as required.
  v2f a;
  a.x = part;
  a.y = 0.0f;
  v2f bones;
  bones.x = 1.0f;
  bones.y = 1.0f;
  v8f acc = {};
  acc = __builtin_amdgcn_wmma_f32_16x16x4_f32(
      /*neg_a=*/false, a, /*neg_b=*/false, bones,
      /*c_mod=*/(short)0, acc, /*reuse_a=*/false, /*reuse_b=*/false);
  float t = acc[0] + acc[1] + acc[2] + acc[3] + acc[4] + acc[5] + acc[6] + acc[7];
  // lane 0 holds sum over rows M=0..7, lane 16 holds rows M=8..15
  float waveSum = __shfl(t, 0, 32) + __shfl(t, 16, 32);
  for (int o = 16; o > 0; o >>= 1) cnt += __shfl_xor(cnt, o, 32);
  if ((threadIdx.x & 31) == 0) {
    atomicAdd((float*)&meta[6], waveSum);
    atomicAdd(&meta[5], (unsigned)cnt);
  }
}

__global__ void ohem_final(const unsigned* __restrict__ meta,
                           float* __restrict__ out) {
  if (threadIdx.x == 0 && blockIdx.x == 0) {
    unsigned k = meta[2];
    if (k == 0u) {
      out[0] = 0.0f;
      return;
    }
    float cutoff = __uint_as_float(meta[3]);
    unsigned ca = meta[5];
    float s = ((const float*)meta)[6];
    // ties at the cutoff value: exactly (k - countAbove) copies contribute
    float total = s + (float)(k - ca) * cutoff;
    out[0] = total / (float)k;
  }
}

extern "C" void kernel_launch(void* const* d_in, const int* in_sizes, int n_in,
                              void* d_out, int out_size, void* d_ws, size_t ws_size,
                              hipStream_t stream) {
  (void)n_in;
  (void)out_size;
  (void)ws_size;
  const float* logits = (const float*)d_in[0];
  const int* labels = (const int*)d_in[1];
  const int P = in_sizes[1]; // 16*512*512 pixels

  float* loss = (float*)d_ws;
  unsigned* meta = (unsigned*)((char*)d_ws + (size_t)P * sizeof(float));
  unsigned* hist = meta + 8;

  ohem_init<<<1, 256, 0, stream>>>(meta);
  dim3 lossGrid(HW_CONST / (256 * 4), NBATCH);
  ohem_loss<<<lossGrid, 256, 0, stream>>>(logits, labels, loss, meta);
  ohem_compute_k<<<1, 32, 0, stream>>>(meta);
  for (int b = 3; b >= 0; --b) {
    ohem_hist<<<2048, 256, 0, stream>>>(loss, P, meta, hist, b);
    ohem_select<<<1, 256, 0, stream>>>(meta, hist, b);
  }
  ohem_sum<<<2048, 256, 0, stream>>>(loss, P, meta);
  ohem_final<<<1, 32, 0, stream>>>(meta, (float*)d_out);
}